// InverseBarkScale_4337916969419
// MI455X (gfx1250) — compile-verified
//
#include <hip/hip_runtime.h>
#include <math.h>

typedef float v2f __attribute__((ext_vector_type(2)));
typedef float v8f __attribute__((ext_vector_type(8)));

#define N_STFT   513
#define N_BARKS  128
#define BATCH    4
#define TIME     512
#define ROWS     (BATCH * TIME)      // 2048
#define FP       528                 // 513 padded up to 33 * 16
#define NB       128
#define LDSS     532                 // LDS spec row stride (mod 64 banks = 20 -> conflict free)
#define LDSD     132                 // LDS diff row stride (mod 64 banks = 4  -> conflict free)
#define MAX_ITER 30
#define LR       0.1f
#define MOMENTUM 0.9f
#define TOL_LOSS 1e-5f
#define TOL_CHG  1e-8f

// ---------------- init kernels ----------------

__global__ __launch_bounds__(256) void ibs_init_spec(const float* __restrict__ spec_init,
                                                     float* __restrict__ spec,
                                                     float* __restrict__ buf) {
    int idx = blockIdx.x * blockDim.x + threadIdx.x;
    if (idx >= ROWS * FP) return;
    int row = idx / FP;
    int f   = idx - row * FP;
    spec[idx] = (f < N_STFT) ? spec_init[row * N_STFT + f] : 0.0f;
    buf[idx]  = 0.0f;
}

// Build k-pair-interleaved copies of the (zero-padded) filterbank:
//   fbp2 : pairs over the frequency (K of GEMM1) dim:  fbp2[(f/2)*NB + kk] = {fb[f][kk], fb[f+1][kk]}
//   fbT2 : pairs over the bark      (K of GEMM2) dim:  fbT2[(kk/2)*FP + f] = {fb[f][kk], fb[f][kk+1]}
// so every WMMA B-operand is one aligned 8-byte load.
__global__ __launch_bounds__(256) void ibs_init_fb(const float* __restrict__ fb_in,
                                                   float* __restrict__ fbp2,
                                                   float* __restrict__ fbT2) {
    int idx = blockIdx.x * blockDim.x + threadIdx.x;
    if (idx >= FP * NB) return;
    int f  = idx / NB;
    int kk = idx - f * NB;
    float v = (f < N_STFT) ? fb_in[f * NB + kk] : 0.0f;
    fbp2[(((f  >> 1) * NB) + kk) * 2 + (f  & 1)] = v;
    fbT2[(((kk >> 1) * FP) + f ) * 2 + (kk & 1)] = v;
}

__global__ void ibs_init_state(float* __restrict__ state) {
    state[0] = INFINITY;  // prev_loss
    state[1] = 0.0f;      // done flag
}

// ---------------- per-iteration fused kernel ----------------
// grid = 128 blocks (one 16-row tile each), block = 256 threads (8 waves)

__global__ __launch_bounds__(256) void ibs_iter(const float* __restrict__ barkspec,
                                                float* __restrict__ spec,
                                                float* __restrict__ buf,
                                                const float2* __restrict__ fbp2,
                                                const float2* __restrict__ fbT2,
                                                float* __restrict__ partials,
                                                const float* __restrict__ state) {
    if (state[1] != 0.0f) return;   // frozen: done flag set on a previous iteration

    __shared__ float lds_spec[16 * LDSS];
    __shared__ float lds_diff[16 * LDSD];
    __shared__ float lds_red[256];

    const int tid     = threadIdx.x;
    const int wave    = tid >> 5;
    const int lane    = tid & 31;
    const int mi      = lane & 15;       // M index for A / N index for B (WMMA layout)
    const int hsel    = lane >> 4;       // half-wave select
    const int koff    = hsel * 2;        // K offset within a 4-wide K step
    const int rowbase = blockIdx.x * 16;

    // ---- stage 16 x FP spec tile into LDS (float4 loads) ----
    for (int i = tid; i < 16 * (FP / 4); i += 256) {
        int r  = i / (FP / 4);
        int c4 = i - r * (FP / 4);
        float4 v = *(const float4*)(spec + (rowbase + r) * FP + c4 * 4);
        float* dst = lds_spec + r * LDSS + c4 * 4;
        dst[0] = v.x; dst[1] = v.y; dst[2] = v.z; dst[3] = v.w;
    }
    __syncthreads();

    // ---- GEMM1: acc = spec_tile (16xFP) x fb (FPx16), wave w owns bark cols [16w,16w+16) ----
    const int nbase = wave * 16;
    // Per-lane base pointers: all lane-variant terms hoisted so the per-step
    // displacement is a compile-time constant -> folds into the load IOFFSET.
    const float*  ap = lds_spec + mi * LDSS + koff;          // A: + k        per step
    const float2* bp = fbp2 + (nbase + mi) + hsel * NB;      // B: + (k/2)*NB per step

    v8f acc = {0.f, 0.f, 0.f, 0.f, 0.f, 0.f, 0.f, 0.f};
    #pragma unroll 4
    for (int k = 0; k < FP; k += 4) {
        v2f a, b;
        a[0] = ap[k];
        a[1] = ap[k + 1];
        float2 bv = bp[(k >> 1) * NB];                       // one b64 load, const offset
        b[0] = bv.x; b[1] = bv.y;
        acc = __builtin_amdgcn_wmma_f32_16x16x4_f32(false, a, false, b,
                                                    (short)0, acc, false, false);
    }

    // ---- diff = m - acc ; stash in LDS ; per-thread sum of squares ----
    const int rowh = rowbase + hsel * 8;
    const int bb   = rowh >> 9;          // batch index (512 rows per batch)
    const int t0   = rowh & 511;
    const float* mp = barkspec + (bb * N_BARKS + nbase + mi) * TIME + t0;
    float ssq = 0.0f;
    #pragma unroll
    for (int r = 0; r < 8; ++r) {
        float d = mp[r] - acc[r];        // acc element r -> row (r + hsel*8), col (nbase+mi)
        ssq += d * d;
        lds_diff[(r + hsel * 8) * LDSD + nbase + mi] = d;
    }
    lds_red[tid] = ssq;
    __syncthreads();

    if (tid == 0) {                      // deterministic fixed-order block reduction
        float s = 0.0f;
        for (int i = 0; i < 256; ++i) s += lds_red[i];
        partials[blockIdx.x] = s;
    }

    // ---- GEMM2 + fused momentum update: grad = (-2/2048) * diff (16x128) x fbT (128x16) ----
    const float* a2p = lds_diff + mi * LDSD + koff;          // A: + k per step
    for (int nt = wave; nt < FP / 16; nt += 8) {
        const int fbase = nt * 16;
        const float2* b2p = fbT2 + (fbase + mi) + hsel * FP; // B: + (k/2)*FP per step
        v8f acc2 = {0.f, 0.f, 0.f, 0.f, 0.f, 0.f, 0.f, 0.f};
        #pragma unroll
        for (int k = 0; k < NB; k += 4) {
            v2f a, b;
            a[0] = a2p[k];
            a[1] = a2p[k + 1];
            float2 bv = b2p[(k >> 1) * FP];                  // one b64 load, const offset
            b[0] = bv.x; b[1] = bv.y;
            acc2 = __builtin_amdgcn_wmma_f32_16x16x4_f32(false, a, false, b,
                                                         (short)0, acc2, false, false);
        }
        #pragma unroll
        for (int r = 0; r < 8; ++r) {
            const int m   = r + hsel * 8;
            const int f   = fbase + mi;
            const int idx = (rowbase + m) * FP + f;
            float g  = acc2[r] * (-2.0f / (float)ROWS);
            float bn = MOMENTUM * buf[idx] + g;
            float sn = lds_spec[m * LDSS + f] - LR * bn;
            spec[idx] = sn > 0.0f ? sn : 0.0f;
            buf[idx]  = bn;
        }
    }
}

// ---------------- stop-condition kernel (1 thread, deterministic sum) ----------------

__global__ void ibs_check(float* __restrict__ state, const float* __restrict__ partials) {
    if (state[1] != 0.0f) return;
    float s = 0.0f;
    for (int i = 0; i < 128; ++i) s += partials[i];
    float loss = s / (float)ROWS;
    float prev = state[0];
    bool stop = (loss < TOL_LOSS) || (fabsf(prev - loss) < TOL_CHG);
    state[0] = loss;
    state[1] = stop ? 1.0f : 0.0f;
}

// ---------------- final transpose (B,T,F) -> (B,F,T) ----------------

__global__ __launch_bounds__(256) void ibs_finalize(const float* __restrict__ spec,
                                                    float* __restrict__ out) {
    int idx = blockIdx.x * blockDim.x + threadIdx.x;
    if (idx >= BATCH * N_STFT * TIME) return;
    int t = idx & 511;
    int f = (idx >> 9) % N_STFT;
    int b = idx / (N_STFT * TIME);
    out[idx] = spec[(b * TIME + t) * FP + f];
}

// ---------------- launch ----------------

extern "C" void kernel_launch(void* const* d_in, const int* in_sizes, int n_in,
                              void* d_out, int out_size, void* d_ws, size_t ws_size,
                              hipStream_t stream) {
    const float* barkspec  = (const float*)d_in[0];  // (4,128,512)
    const float* fb        = (const float*)d_in[1];  // (513,128)
    const float* spec_init = (const float*)d_in[2];  // (4,512,513)

    float* ws       = (float*)d_ws;
    float* spec     = ws;                      // ROWS*FP
    float* buf      = spec + ROWS * FP;        // ROWS*FP
    float* fbp2     = buf  + ROWS * FP;        // FP*NB (pair-interleaved)
    float* fbT2     = fbp2 + FP * NB;          // NB*FP (pair-interleaved, transposed)
    float* partials = fbT2 + NB * FP;          // 128
    float* state    = partials + 128;          // 2

    ibs_init_spec<<<(ROWS * FP + 255) / 256, 256, 0, stream>>>(spec_init, spec, buf);
    ibs_init_fb<<<(FP * NB + 255) / 256, 256, 0, stream>>>(fb, fbp2, fbT2);
    ibs_init_state<<<1, 1, 0, stream>>>(state);

    for (int it = 0; it < MAX_ITER; ++it) {
        ibs_iter<<<ROWS / 16, 256, 0, stream>>>(barkspec, spec, buf,
                                                (const float2*)fbp2, (const float2*)fbT2,
                                                partials, state);
        if (it < MAX_ITER - 1)
            ibs_check<<<1, 1, 0, stream>>>(state, partials);
    }

    ibs_finalize<<<(BATCH * N_STFT * TIME + 255) / 256, 256, 0, stream>>>(spec, (float*)d_out);
}